// Loss_CNN_VAE_40200893891226
// MI455X (gfx1250) — compile-verified
//
#include <hip/hip_runtime.h>
#include <hip/hip_bf16.h>

typedef float v2f __attribute__((ext_vector_type(2)));
typedef float v8f __attribute__((ext_vector_type(8)));

#define KL_TOT 128          // K*L = 16*8
#define HW_TOT 40000        // H*W = 200*200
#define SCALE  5000.0f
#define MEAN_DENOM 1920000.0f   // K*C*H*W = 16*3*200*200

// Phase 1: one block = one 16-wide HW tile across all 128 KL rows (8 waves x 16 rows).
// Each wave computes its 16x16 prob tile with a single V_WMMA_F32_16X16X4_F32
// (K=3 zero-padded to 4, SCALE folded into the A operand), then streams the
// matching x values and accumulates sum(|x-prob|^2). Block partial -> d_ws.
__global__ __launch_bounds__(256) void prob_mse_kernel(
    const float* __restrict__ x,     // [128][3][40000]
    const float* __restrict__ beta,  // [128][3]
    const float* __restrict__ Ain,   // [40000][3]
    float* __restrict__ partial)     // [gridDim.x]
{
    const int tid   = threadIdx.x;
    const int lane  = tid & 31;
    const int wave  = tid >> 5;          // 0..7 -> M tile of 16 KL rows
    const int mbase = wave * 16;
    const int hw0   = blockIdx.x * 16;

    const int  m  = lane & 15;
    const bool hi = lane >= 16;

    // A-matrix (16x4 f32): lanes 0-15 -> {K0,K1}, lanes 16-31 -> {K2,K3=0}
    const float* bp = beta + (mbase + m) * 3;
    v2f a;
    a.x = (hi ? bp[2] : bp[0]) * SCALE;
    a.y = (hi ? 0.0f  : bp[1]) * SCALE;

    // B-matrix (4x16 f32): lanes 0-15 -> rows K0,K1 at col m; lanes 16-31 -> K2,K3=0
    const float* ap = Ain + (hw0 + m) * 3;
    v2f b;
    b.x = hi ? ap[2] : ap[0];
    b.y = hi ? 0.0f  : ap[1];

    v8f c = {};
    // D = A x B + 0 ; emits v_wmma_f32_16x16x4_f32
    c = __builtin_amdgcn_wmma_f32_16x16x4_f32(false, a, false, b, (short)0, c,
                                              false, false);
    // c[r] = prob[mbase + r + (hi?8:0)][hw0 + m]

    const int col = hw0 + m;
    const int klb = mbase + (hi ? 8 : 0);
    float acc = 0.0f;
    #pragma unroll
    for (int r = 0; r < 8; ++r) {
        const float p = c[r];
        const float* xp = x + (size_t)(klb + r) * 3 * HW_TOT + col;
        #pragma unroll
        for (int ch = 0; ch < 3; ++ch) {
            float d = fabsf(xp[(size_t)ch * HW_TOT] - p);
            acc = fmaf(d, d, acc);
        }
    }

    __shared__ float red[256];
    red[tid] = acc;
    __syncthreads();
    #pragma unroll
    for (int s = 128; s > 0; s >>= 1) {
        if (tid < s) red[tid] += red[tid + s];
        __syncthreads();
    }
    if (tid == 0) partial[blockIdx.x] = red[0];
}

// Phase 2: deterministic single-block reduction of the 2500 block partials,
// then apply the mean denominator. Bit-identical across graph replays.
__global__ __launch_bounds__(256) void reduce_kernel(
    const float* __restrict__ partial, int n, float* __restrict__ out)
{
    __shared__ float red[256];
    float acc = 0.0f;
    for (int i = threadIdx.x; i < n; i += 256) acc += partial[i];
    red[threadIdx.x] = acc;
    __syncthreads();
    #pragma unroll
    for (int s = 128; s > 0; s >>= 1) {
        if (threadIdx.x < s) red[threadIdx.x] += red[threadIdx.x + s];
        __syncthreads();
    }
    if (threadIdx.x == 0) out[0] = red[0] * (1.0f / MEAN_DENOM);
}

extern "C" void kernel_launch(void* const* d_in, const int* in_sizes, int n_in,
                              void* d_out, int out_size, void* d_ws, size_t ws_size,
                              hipStream_t stream) {
    const float* x    = (const float*)d_in[0];   // [16,8,3,200,200]
    const float* beta = (const float*)d_in[1];   // [16,8,3]
    const float* Ain  = (const float*)d_in[2];   // [200,200,3]
    float* partial = (float*)d_ws;               // 2500 floats scratch

    const int nblocks = HW_TOT / 16;             // 2500 exact
    prob_mse_kernel<<<nblocks, 256, 0, stream>>>(x, beta, Ain, partial);
    reduce_kernel<<<1, 256, 0, stream>>>(partial, nblocks, (float*)d_out);
}